// VectorQuantizer_76321568850394
// MI455X (gfx1250) — compile-verified
//
#include <hip/hip_runtime.h>
#include <math.h>

typedef float v2f __attribute__((ext_vector_type(2)));
typedef float v8f __attribute__((ext_vector_type(8)));

#define NUM_CODES 512
#define EDIM      64
#define CB_STRIDE 68            // padded LDS row stride (floats): stride%64 banks == 4 -> conflict-free B reads
#define HW        4096          // H*W
#define CHW       (EDIM * HW)   // per-batch stride in x
#define NROWS     131072        // B*H*W
#define WAVES_PER_WG 8
#define NBLOCKS   512
#define QELEMS    8388608       // 32*64*64*64

// ---------------- init: zero the code-usage counters (replayed every graph launch) ----------------
__global__ void vq_init(unsigned int* __restrict__ counts) {
    counts[threadIdx.x] = 0u;
}

// ---------------- main: fused distance-GEMM (WMMA fp32) + argmin + quantize + partial loss -------
__global__ __launch_bounds__(256) void vq_main(const float* __restrict__ x,
                                               const float* __restrict__ emb,
                                               float* __restrict__ out,
                                               unsigned int* __restrict__ counts,
                                               float* __restrict__ partial) {
    __shared__ float s_cb[NUM_CODES * CB_STRIDE];   // 139264 B codebook (padded rows)
    __shared__ float s_norm[NUM_CODES];             // ||e_k||^2
    __shared__ int   s_idx[WAVES_PER_WG][32];       // per-wave winning code indices
    __shared__ float s_red[256];                    // loss reduction scratch

    const int tid = threadIdx.x;

    // cooperative codebook load (float4 granularity), padded rows in LDS
    for (int i = tid; i < NUM_CODES * (EDIM / 4); i += 256) {
        const int k  = i >> 4;     // 16 float4 per code row
        const int c4 = i & 15;
        const float4 v = ((const float4*)emb)[i];
        *(float4*)&s_cb[k * CB_STRIDE + c4 * 4] = v;
    }
    __syncthreads();

    // per-code squared norms
    for (int k = tid; k < NUM_CODES; k += 256) {
        float s = 0.f;
        #pragma unroll 8
        for (int c = 0; c < EDIM; ++c) { const float e = s_cb[k * CB_STRIDE + c]; s += e * e; }
        s_norm[k] = s;
    }
    __syncthreads();

    const int wv   = tid >> 5;
    const int lane = tid & 31;
    const int task = blockIdx.x * WAVES_PER_WG + wv;  // 0..4095, 32 rows each
    const int n0   = task * 32;
    const int b    = n0 >> 12;            // 4096 pixels per batch image
    const int hw0  = n0 & (HW - 1);       // 32-aligned, same (b) for all 32 rows
    const int xbase = b * CHW + hw0;

    const int lan   = lane & 15;          // position within 16-lane half
    const int khalf = (lane >> 4) * 2;    // K sub-offset for A/B layout halves

    // A tiles: 16x4 fp32 layout -> lanes0-15: K={4j,4j+1}, lanes16-31: K={4j+2,4j+3}
    v2f a_lo[16], a_hi[16];               // rows hw0+0..15 and hw0+16..31
    #pragma unroll
    for (int j = 0; j < 16; ++j) {
        const int c0 = 4 * j + khalf;
        a_lo[j].x = x[xbase + (c0    ) * HW + lan];
        a_lo[j].y = x[xbase + (c0 + 1) * HW + lan];
        a_hi[j].x = x[xbase + (c0    ) * HW + 16 + lan];
        a_hi[j].y = x[xbase + (c0 + 1) * HW + 16 + lan];
    }

    float bs0[8], bs1[8];
    int   bi0[8], bi1[8];
    #pragma unroll
    for (int v = 0; v < 8; ++v) { bs0[v] = 3.4e38f; bs1[v] = 3.4e38f; bi0[v] = 0; bi1[v] = 0; }

    // sweep all 32 code tiles of 16; 32 v_wmma_f32_16x16x4_f32 per iteration
    for (int tt = 0; tt < 32; ++tt) {
        v8f acc0 = {}; v8f acc1 = {};
        const int nb = tt * 16;
        const float* brow = &s_cb[(nb + lan) * CB_STRIDE + khalf];   // 8B-aligned
        #pragma unroll
        for (int j = 0; j < 16; ++j) {
            const v2f bb = *(const v2f*)(brow + 4 * j);
            acc0 = __builtin_amdgcn_wmma_f32_16x16x4_f32(false, a_lo[j], false, bb,
                                                         (short)0, acc0, false, false);
            acc1 = __builtin_amdgcn_wmma_f32_16x16x4_f32(false, a_hi[j], false, bb,
                                                         (short)0, acc1, false, false);
        }
        const int   nidx = nb + lan;
        const float en   = s_norm[nidx];
        #pragma unroll
        for (int v = 0; v < 8; ++v) {
            const float s0 = en - 2.0f * acc0[v];   // ||e||^2 - 2 x.e  (||x||^2 is row-constant)
            if (s0 < bs0[v]) { bs0[v] = s0; bi0[v] = nidx; }
            const float s1 = en - 2.0f * acc1[v];
            if (s1 < bs1[v]) { bs1[v] = s1; bi1[v] = nidx; }
        }
    }

    // argmin across the 16 lanes of each half (rows v in lanes0-15, rows 8+v in lanes16-31)
    #pragma unroll
    for (int v = 0; v < 8; ++v) {
        #pragma unroll
        for (int off = 1; off < 16; off <<= 1) {
            float os = __shfl_xor(bs0[v], off, 32);
            int   oi = __shfl_xor(bi0[v], off, 32);
            if (os < bs0[v] || (os == bs0[v] && oi < bi0[v])) { bs0[v] = os; bi0[v] = oi; }
            os = __shfl_xor(bs1[v], off, 32);
            oi = __shfl_xor(bi1[v], off, 32);
            if (os < bs1[v] || (os == bs1[v] && oi < bi1[v])) { bs1[v] = os; bi1[v] = oi; }
        }
    }

    if (lane == 0) {
        #pragma unroll
        for (int v = 0; v < 8; ++v) { s_idx[wv][v]      = bi0[v]; s_idx[wv][16 + v] = bi1[v]; }
    } else if (lane == 16) {
        #pragma unroll
        for (int v = 0; v < 8; ++v) { s_idx[wv][8 + v]  = bi0[v]; s_idx[wv][24 + v] = bi1[v]; }
    }
    __syncthreads();

    // cooperative write-out: lane L owns row hw0+L; fully coalesced 128B stores per channel
    const int myidx = s_idx[wv][lane];
    const float* qrow = &s_cb[myidx * CB_STRIDE];
    float lsum = 0.f;
    const int obase = xbase + lane;
    #pragma unroll 4
    for (int c = 0; c < EDIM; ++c) {
        const float q  = qrow[c];
        const float xv = x[obase + c * HW];
        const float d  = q - xv;
        lsum += d * d;
        out[obase + c * HW] = q;
    }
    atomicAdd(&counts[myidx], 1u);   // integer atomics: deterministic

    // fixed-order workgroup loss reduction -> per-block partial
    s_red[tid] = lsum;
    __syncthreads();
    for (int s = 128; s > 0; s >>= 1) {
        if (tid < s) s_red[tid] += s_red[tid + s];
        __syncthreads();
    }
    if (tid == 0) partial[blockIdx.x] = s_red[0];
}

// ---------------- finalize: loss + perplexity scalars ----------------
__global__ __launch_bounds__(512) void vq_finalize(const unsigned int* __restrict__ counts,
                                                   const float* __restrict__ partial,
                                                   float* __restrict__ out) {
    __shared__ float s1[512];   // entropy terms
    __shared__ float s2[512];   // loss partials
    const int t = threadIdx.x;
    const float cnt = (float)counts[t];
    const float avg = cnt * (1.0f / (float)NROWS);
    s1[t] = avg * logf(avg + 1e-10f);
    s2[t] = partial[t];          // NBLOCKS == 512
    __syncthreads();
    for (int s = 256; s > 0; s >>= 1) {
        if (t < s) { s1[t] += s1[t + s]; s2[t] += s2[t + s]; }
        __syncthreads();
    }
    if (t == 0) {
        out[QELEMS]     = 1.25f * s2[0] / (float)QELEMS;  // q_loss + 0.25*e_loss = 1.25*MSE
        out[QELEMS + 1] = expf(-s1[0]);
    }
}

extern "C" void kernel_launch(void* const* d_in, const int* in_sizes, int n_in,
                              void* d_out, int out_size, void* d_ws, size_t ws_size,
                              hipStream_t stream) {
    const float* x   = (const float*)d_in[0];
    const float* emb = (const float*)d_in[1];
    float* out = (float*)d_out;
    unsigned int* counts = (unsigned int*)d_ws;
    float* partial = (float*)((char*)d_ws + NUM_CODES * sizeof(unsigned int));

    vq_init<<<1, NUM_CODES, 0, stream>>>(counts);
    vq_main<<<NBLOCKS, 256, 0, stream>>>(x, emb, out, counts, partial);
    vq_finalize<<<1, 512, 0, stream>>>(counts, partial, out);
}